// GAT_43559558316089
// MI455X (gfx1250) — compile-verified
//
#include <hip/hip_runtime.h>

// Problem constants (match reference).
#define NNODES 50000
#define NEDGES 800000
#define FIN    128
#define H1     8      // heads, layer 1
#define HID1   8      // channels per head, layer 1
#define C1     64     // H1*HID1
#define C2     32     // layer-2 output channels (1 head)
#define NGRAPH 64
#define NEG_SLOPE 0.2f

typedef __attribute__((ext_vector_type(2))) float v2f;
typedef __attribute__((ext_vector_type(8))) float v8f;

// ---- order-preserving float<->uint transform for atomic max on signed floats
__device__ __forceinline__ unsigned f2ord(float f) {
  unsigned u = __float_as_uint(f);
  return (u & 0x80000000u) ? ~u : (u | 0x80000000u);
}
__device__ __forceinline__ float ord2f(unsigned u) {
  return (u & 0x80000000u) ? __uint_as_float(u & 0x7fffffffu) : __uint_as_float(~u);
}
#define ORD_NEG_INF 0x007FFFFFu  // f2ord(-inf)

// ---------------------------------------------------------------------------
// fill helper
__global__ void fill_u32(unsigned* __restrict__ p, unsigned v, int n) {
  int i = blockIdx.x * blockDim.x + threadIdx.x;
  if (i < n) p[i] = v;
}

// ---------------------------------------------------------------------------
// f32 WMMA GEMM: C[nrows x NDIM] = A[nrows x KDIM] * B[KDIM x NDIM]
// One 16x16 output tile per wave; blockIdx.x = row tile, wave id = col tile.
// Uses V_WMMA_F32_16X16X4_F32 (K=4 per issue), f32 precision parity with ref.
template <int KDIM, int NDIM>
__global__ void wmma_gemm_f32(const float* __restrict__ A,
                              const float* __restrict__ B,
                              float* __restrict__ C) {
  const int wave = threadIdx.x >> 5;
  const int lane = threadIdx.x & 31;
  const int row0 = blockIdx.x * 16;
  const int col0 = wave * 16;
  const int m  = lane & 15;         // row (A) / col (B,C) within tile
  const int kk = (lane >> 4) << 1;  // lanes 16-31 hold K+2,K+3

  v8f acc = {};
  for (int k = 0; k < KDIM; k += 4) {
    v2f a, b;
    const float* ap = A + (size_t)(row0 + m) * KDIM + (k + kk);
    a.x = ap[0];
    a.y = ap[1];
    const float* bp = B + (size_t)(k + kk) * NDIM + col0 + m;
    b.x = bp[0];
    b.y = bp[NDIM];
    acc = __builtin_amdgcn_wmma_f32_16x16x4_f32(
        /*neg_a=*/false, a, /*neg_b=*/false, b,
        /*c_mod=*/(short)0, acc, /*reuse_a=*/false, /*reuse_b=*/false);
  }
  // C/D layout: VGPR i holds row M = i + (lane>=16 ? 8 : 0), col = lane&15
  const int rbase = row0 + ((lane >> 4) << 3);
  for (int i = 0; i < 8; ++i)
    C[(size_t)(rbase + i) * NDIM + col0 + m] = acc[i];
}

// ---------------------------------------------------------------------------
// Per-node attention halves: als[n,h] = <feat[n,h,:], asrc[h,:]>, same for ald.
template <int H, int CPH>
__global__ void node_att(const float* __restrict__ feat,
                         const float* __restrict__ asrc,
                         const float* __restrict__ adst,
                         float* __restrict__ als, float* __restrict__ ald,
                         int total) {
  int idx = blockIdx.x * blockDim.x + threadIdx.x;
  if (idx >= total) return;
  int n = idx / H, hd = idx % H;
  const float* f = feat + (size_t)n * (H * CPH) + hd * CPH;
  float ss = 0.f, sd = 0.f;
  for (int c = 0; c < CPH; ++c) {
    ss += f[c] * asrc[hd * CPH + c];
    sd += f[c] * adst[hd * CPH + c];
  }
  als[idx] = ss;
  ald[idx] = sd;
}

// ---------------------------------------------------------------------------
// Edge passes. Edge list = E real edges + N implicit self-loops.
__device__ __forceinline__ void edge_ends(const int* __restrict__ ei, int edge,
                                          int& s, int& d) {
  if (edge < NEDGES) { s = ei[edge]; d = ei[NEDGES + edge]; }
  else               { s = d = edge - NEDGES; }
}

template <int H>
__global__ void edge_seg_max(const int* __restrict__ ei,
                             const float* __restrict__ als,
                             const float* __restrict__ ald,
                             unsigned* __restrict__ mo, int total) {
  int idx = blockIdx.x * blockDim.x + threadIdx.x;
  if (idx >= total) return;
  int edge = idx / H, hd = idx % H;
  int s, d; edge_ends(ei, edge, s, d);
  float v = als[s * H + hd] + ald[d * H + hd];
  v = v > 0.f ? v : NEG_SLOPE * v;
  atomicMax(&mo[d * H + hd], f2ord(v));
}

template <int H>
__global__ void edge_seg_den(const int* __restrict__ ei,
                             const float* __restrict__ als,
                             const float* __restrict__ ald,
                             const unsigned* __restrict__ mo,
                             float* __restrict__ den, int total) {
  int idx = blockIdx.x * blockDim.x + threadIdx.x;
  if (idx >= total) return;
  int edge = idx / H, hd = idx % H;
  int s, d; edge_ends(ei, edge, s, d);
  float v = als[s * H + hd] + ald[d * H + hd];
  v = v > 0.f ? v : NEG_SLOPE * v;
  atomicAdd(&den[d * H + hd], expf(v - ord2f(mo[d * H + hd])));
}

template <int C, int CPH, int H>
__global__ void edge_scatter(const int* __restrict__ ei,
                             const float* __restrict__ feat,
                             const float* __restrict__ als,
                             const float* __restrict__ ald,
                             const unsigned* __restrict__ mo,
                             const float* __restrict__ den,
                             float* __restrict__ outb, int total) {
  int idx = blockIdx.x * blockDim.x + threadIdx.x;
  if (idx >= total) return;
  int edge = idx / C, c = idx % C, hd = c / CPH;
  int s, d; edge_ends(ei, edge, s, d);
  float v = als[s * H + hd] + ald[d * H + hd];
  v = v > 0.f ? v : NEG_SLOPE * v;
  float alpha = expf(v - ord2f(mo[d * H + hd])) / (den[d * H + hd] + 1e-16f);
  atomicAdd(&outb[(size_t)d * C + c], feat[(size_t)s * C + c] * alpha);
}

// ---------------------------------------------------------------------------
__global__ void bias_elu(float* __restrict__ o, const float* __restrict__ b,
                         int total) {
  int idx = blockIdx.x * blockDim.x + threadIdx.x;
  if (idx >= total) return;
  float z = o[idx] + b[idx & (C1 - 1)];
  o[idx] = z > 0.f ? z : expm1f(z);
}

__global__ void pool_sum(const float* __restrict__ o2, const float* __restrict__ b2,
                         const int* __restrict__ batch, float* __restrict__ gsum,
                         int total) {
  int idx = blockIdx.x * blockDim.x + threadIdx.x;
  if (idx >= total) return;
  int n = idx >> 5, c = idx & (C2 - 1);
  atomicAdd(&gsum[batch[n] * C2 + c], o2[idx] + b2[c]);
}

__global__ void pool_cnt(const int* __restrict__ batch, float* __restrict__ gcnt,
                         int n) {
  int i = blockIdx.x * blockDim.x + threadIdx.x;
  if (i < n) atomicAdd(&gcnt[batch[i]], 1.f);
}

__global__ void final_linear(const float* __restrict__ gsum,
                             const float* __restrict__ gcnt,
                             const float* __restrict__ Wlin,
                             const float* __restrict__ blin,
                             float* __restrict__ out) {
  int idx = threadIdx.x;
  if (idx >= NGRAPH * 2) return;
  int g = idx >> 1, o = idx & 1;
  float inv = 1.f / fmaxf(gcnt[g], 1.f);
  float s = blin[o];
  for (int c = 0; c < C2; ++c) s += (gsum[g * C2 + c] * inv) * Wlin[c * 2 + o];
  out[idx] = s;
}

// ---------------------------------------------------------------------------
static inline int gridFor(int n, int blk) { return (n + blk - 1) / blk; }

extern "C" void kernel_launch(void* const* d_in, const int* in_sizes, int n_in,
                              void* d_out, int out_size, void* d_ws, size_t ws_size,
                              hipStream_t stream) {
  const float* x    = (const float*)d_in[0];
  const int*   ei   = (const int*)d_in[1];
  const int*   batch= (const int*)d_in[2];
  const float* W1   = (const float*)d_in[3];
  const float* a1s  = (const float*)d_in[4];
  const float* a1d  = (const float*)d_in[5];
  const float* b1   = (const float*)d_in[6];
  const float* W2   = (const float*)d_in[7];
  const float* a2s  = (const float*)d_in[8];
  const float* a2d  = (const float*)d_in[9];
  const float* b2   = (const float*)d_in[10];
  const float* Wlin = (const float*)d_in[11];
  const float* blin = (const float*)d_in[12];
  float* out = (float*)d_out;

  // workspace carve-up (all 4-byte typed)
  float* p = (float*)d_ws;
  float*    h1   = p; p += (size_t)NNODES * C1;
  float*    o1   = p; p += (size_t)NNODES * C1;
  float*    als1 = p; p += (size_t)NNODES * H1;
  float*    ald1 = p; p += (size_t)NNODES * H1;
  unsigned* m1   = (unsigned*)p; p += (size_t)NNODES * H1;
  float*    den1 = p; p += (size_t)NNODES * H1;
  float*    h2   = p; p += (size_t)NNODES * C2;
  float*    o2   = p; p += (size_t)NNODES * C2;
  float*    als2 = p; p += NNODES;
  float*    ald2 = p; p += NNODES;
  unsigned* m2   = (unsigned*)p; p += NNODES;
  float*    den2 = p; p += NNODES;
  float*    gsum = p; p += NGRAPH * C2;
  float*    gcnt = p; p += NGRAPH;

  const int EP = NEDGES + NNODES;  // edges incl. self-loops

  // init accumulators
  fill_u32<<<gridFor(NNODES * H1, 256), 256, 0, stream>>>(m1, ORD_NEG_INF, NNODES * H1);
  fill_u32<<<gridFor(NNODES * H1, 256), 256, 0, stream>>>((unsigned*)den1, 0u, NNODES * H1);
  fill_u32<<<gridFor(NNODES * C1, 256), 256, 0, stream>>>((unsigned*)o1, 0u, NNODES * C1);
  fill_u32<<<gridFor(NNODES, 256), 256, 0, stream>>>(m2, ORD_NEG_INF, NNODES);
  fill_u32<<<gridFor(NNODES, 256), 256, 0, stream>>>((unsigned*)den2, 0u, NNODES);
  fill_u32<<<gridFor(NNODES * C2, 256), 256, 0, stream>>>((unsigned*)o2, 0u, NNODES * C2);
  fill_u32<<<1, 256, 0, stream>>>((unsigned*)gsum, 0u, NGRAPH * C2);
  fill_u32<<<1, 64, 0, stream>>>((unsigned*)gcnt, 0u, NGRAPH);

  // ---- layer 1
  // h1 = x @ W1  (50000x128 @ 128x64), 4 col tiles -> 4 waves/block
  wmma_gemm_f32<FIN, C1><<<NNODES / 16, 128, 0, stream>>>(x, W1, h1);
  node_att<H1, HID1><<<gridFor(NNODES * H1, 256), 256, 0, stream>>>(
      h1, a1s, a1d, als1, ald1, NNODES * H1);
  edge_seg_max<H1><<<gridFor(EP * H1, 256), 256, 0, stream>>>(ei, als1, ald1, m1, EP * H1);
  edge_seg_den<H1><<<gridFor(EP * H1, 256), 256, 0, stream>>>(ei, als1, ald1, m1, den1, EP * H1);
  edge_scatter<C1, HID1, H1><<<gridFor(EP * C1, 256), 256, 0, stream>>>(
      ei, h1, als1, ald1, m1, den1, o1, EP * C1);
  bias_elu<<<gridFor(NNODES * C1, 256), 256, 0, stream>>>(o1, b1, NNODES * C1);

  // ---- layer 2
  // h2 = elu(o1) @ W2  (50000x64 @ 64x32), 2 col tiles -> 2 waves/block
  wmma_gemm_f32<C1, C2><<<NNODES / 16, 64, 0, stream>>>(o1, W2, h2);
  node_att<1, C2><<<gridFor(NNODES, 256), 256, 0, stream>>>(
      h2, a2s, a2d, als2, ald2, NNODES);
  edge_seg_max<1><<<gridFor(EP, 256), 256, 0, stream>>>(ei, als2, ald2, m2, EP);
  edge_seg_den<1><<<gridFor(EP, 256), 256, 0, stream>>>(ei, als2, ald2, m2, den2, EP);
  edge_scatter<C2, C2, 1><<<gridFor(EP * C2, 256), 256, 0, stream>>>(
      ei, h2, als2, ald2, m2, den2, o2, EP * C2);

  // ---- pooling + linear head
  pool_sum<<<gridFor(NNODES * C2, 256), 256, 0, stream>>>(o2, b2, batch, gsum, NNODES * C2);
  pool_cnt<<<gridFor(NNODES, 256), 256, 0, stream>>>(batch, gcnt, NNODES);
  final_linear<<<1, 128, 0, stream>>>(gsum, gcnt, Wlin, blin, out);
}